// SineGen_36764920054572
// MI455X (gfx1250) — compile-verified
//
#include <hip/hip_runtime.h>
#include <stdint.h>

// ---------------- problem constants (match reference) ----------------
#define B_    16
#define T_    1000
#define UPP_  256
#define H_    9
#define TUP_  (T_ * UPP_)            // 256000
#define NFRAMES_ (B_ * T_)           // 16000 frames; frame = b*1000 + k
#define FSZ_  (UPP_ * H_)            // 2304 elements per frame (contiguous)
#define SR_INV (1.0f / 44100.0f)
#define SINE_AMP 0.1f
#define NOISE_STD 0.003f

// ---------------- tiling for the streaming kernel --------------------
#define FPG_   4                     // frames per async group (36 KB tile)
#define GRPS_  4                     // groups per block
#define FPB_   (FPG_ * GRPS_)        // 16 frames per block
#define NBLK_  (NFRAMES_ / FPB_)     // 1000 blocks, exact
#define GELEMS_ (FPG_ * FSZ_)        // 9216 floats per group
#define GBYTES_ (GELEMS_ * 4)        // 36864 bytes per group
#define ACHUNKS_ 9                   // async b128 chunks per lane per group

typedef float v4f  __attribute__((ext_vector_type(4)));
typedef float v2f  __attribute__((ext_vector_type(2)));
typedef float v4fu __attribute__((ext_vector_type(4), aligned(4)));

#if __has_builtin(__builtin_amdgcn_sinf)
// llvm.amdgcn.sin takes input in revolutions: sin(2*pi*x)
#define FAST_SIN_REV(x) __builtin_amdgcn_sinf(x)
#else
#define FAST_SIN_REV(x) __sinf((x) * 6.28318530717958647692f)
#endif

// CDNA5 async global->LDS copy (gfx1250). VDST VGPR carries the LDS byte
// address; GVS mode: mem_addr = SGPR64 + VGPR32 offset. Tracked by ASYNCcnt.
__device__ __forceinline__ void stage_async_b128(unsigned lds_addr,
                                                 uint64_t gbase,
                                                 unsigned byte_off) {
  asm volatile("global_load_async_to_lds_b128 %0, %1, %2"
               :
               : "v"(lds_addr), "v"(byte_off), "s"(gbase)
               : "memory");
}
__device__ __forceinline__ void wait_async_le9() {
  asm volatile("s_wait_asynccnt 9" ::: "memory");
}
__device__ __forceinline__ void wait_async_le0() {
  asm volatile("s_wait_asynccnt 0" ::: "memory");
}

// ---------------------------------------------------------------------
// Kernel A: per-batch frame table
//   FP[b][k].x = (sum_{j<k} f0[b][j]) * UPP / SR   (exclusive prefix, revs)
//   FP[b][k].y = f0[b][k] / SR                      (per-sample increment)
// ---------------------------------------------------------------------
__global__ __launch_bounds__(256) void sinegen_scan(const float* __restrict__ f0,
                                                    v2f* __restrict__ FP) {
  __shared__ float sums[256];
  const int b   = blockIdx.x;
  const int tid = threadIdx.x;
  const int jb  = tid * 4;

  float v[4];
  float s = 0.0f;
#pragma unroll
  for (int u = 0; u < 4; ++u) {
    const int j = jb + u;
    const float x = f0[b * T_ + (j < T_ ? j : T_ - 1)];  // clamped, branchless
    v[u] = (j < T_) ? x : 0.0f;
    s += v[u];
  }
  sums[tid] = s;
  __syncthreads();

#pragma unroll
  for (int off = 1; off < 256; off <<= 1) {
    const float add = (tid >= off) ? sums[tid - off] : 0.0f;
    __syncthreads();
    sums[tid] += add;
    __syncthreads();
  }

  float run = sums[tid] - s;
  const float c = (float)UPP_ * SR_INV;
#pragma unroll
  for (int u = 0; u < 4; ++u) {
    const int j = jb + u;
    if (j < T_) {
      v2f pv;
      pv.x = run * c;
      pv.y = v[u] * SR_INV;
      FP[b * T_ + j] = pv;
    }
    run += v[u];
  }
}

// ---------------------------------------------------------------------
// Kernel B: frame-mapped streaming sine generation.
// Block handles 16 frames; thread tid = time sample i within each frame and
// computes all 9 harmonics:  elem = frame*2304 + tid*9 + h.
//   phi   = FP.x + FP.y*(tid+1)                (per-lane, 1 fma)
//   ph_h  = phi*(h+1) + rand_ini[b][h]         (uniform scalars, imm mults)
//   out   = amp*sin(2*pi*frac(ph)) + namp*noise
// Noise streamed via double-buffered async global->LDS groups of 4 frames.
// ---------------------------------------------------------------------
__global__ __launch_bounds__(256) void sinegen_main(
    const v2f* __restrict__ FP, const float* __restrict__ rand_ini,
    const float* __restrict__ noise, float* __restrict__ out) {
  __shared__ float stage[2][GELEMS_];  // 2 x 36 KB

  const unsigned tid    = threadIdx.x;
  const unsigned frameB = blockIdx.x * (unsigned)FPB_;
  const uint64_t nbase  = (uint64_t)noise;
  const unsigned ldsA   = (unsigned)(uintptr_t)&stage[0][0] + tid * 16u;
  const unsigned ldsB   = (unsigned)(uintptr_t)&stage[1][0] + tid * 16u;
  const float    tid1f  = (float)(tid + 1u);

  // ---- prime: stage group 0 into buffer 0 (9 coalesced async b128 / lane)
#pragma unroll
  for (int j = 0; j < ACHUNKS_; ++j)
    stage_async_b128(ldsA + (unsigned)j * 4096u, nbase,
                     frameB * (unsigned)(FSZ_ * 4) + (unsigned)j * 4096u + tid * 16u);

  for (int g = 0; g < GRPS_; ++g) {
    if (g + 1 < GRPS_) {
      const unsigned nlds = ((g + 1) & 1) ? ldsB : ldsA;
      const unsigned goff = (frameB + (unsigned)(g + 1) * FPG_) * (unsigned)(FSZ_ * 4);
#pragma unroll
      for (int j = 0; j < ACHUNKS_; ++j)
        stage_async_b128(nlds + (unsigned)j * 4096u, nbase,
                         goff + (unsigned)j * 4096u + tid * 16u);
      wait_async_le9();  // own group-g loads have landed (next group in flight)
    } else {
      wait_async_le0();
    }
    __syncthreads();  // buffer g complete across all waves

    const float* buf = &stage[g & 1][0];
#pragma unroll
    for (int f = 0; f < FPG_; ++f) {
      const unsigned frame = frameB + (unsigned)g * FPG_ + (unsigned)f;  // uniform
      const v2f pf = FP[frame];                     // uniform -> s_load_b64
      const unsigned bb = frame / 1000u;            // uniform batch index
      const float* ri = rand_ini + bb * (unsigned)H_;

      const float phi = fmaf(pf.y, tid1f, pf.x);    // base phase (revolutions)
      const bool voiced = pf.y > 0.0f;              // uniform per frame
      const float amp  = voiced ? SINE_AMP : 0.0f;
      const float namp = voiced ? NOISE_STD : (SINE_AMP / 3.0f);

      const float* np = buf + f * FSZ_ + tid * (unsigned)H_;
      const v4fu n0 = *(const v4fu*)np;             // 4B-aligned b128
      const v4fu n1 = *(const v4fu*)(np + 4);
      const float n8 = np[8];
      const float nz[9] = {n0.x, n0.y, n0.z, n0.w, n1.x, n1.y, n1.z, n1.w, n8};

      float res[9];
#pragma unroll
      for (int h = 0; h < H_; ++h) {
        const float ph  = fmaf(phi, (float)(h + 1), ri[h]);  // imm mult, sgpr add
        const float red = ph - rintf(ph);                    // frac -> [-.5,.5] revs
        res[h] = fmaf(amp, FAST_SIN_REV(red), namp * nz[h]);
      }

      float* op = out + (size_t)frame * (size_t)FSZ_ + (size_t)tid * (size_t)H_;
      const v4fu r0 = {res[0], res[1], res[2], res[3]};
      const v4fu r1 = {res[4], res[5], res[6], res[7]};
      __builtin_nontemporal_store(r0, (v4fu*)op);
      __builtin_nontemporal_store(r1, (v4fu*)(op + 4));
      __builtin_nontemporal_store(res[8], op + 8);
    }
    __syncthreads();  // everyone done reading buf[g&1] before it is re-staged
  }
}

// ---------------------------------------------------------------------
extern "C" void kernel_launch(void* const* d_in, const int* in_sizes, int n_in,
                              void* d_out, int out_size, void* d_ws, size_t ws_size,
                              hipStream_t stream) {
  const float* f0       = (const float*)d_in[0];  // (16, 1000)
  const float* rand_ini = (const float*)d_in[1];  // (16, 9)
  const float* noise    = (const float*)d_in[2];  // (16, 256000, 9)
  // d_in[3] = upp (scalar 256), baked into the kernels

  v2f*   FP  = (v2f*)d_ws;     // 16*1000 float2 = 128 KB scratch, fully rewritten
  float* out = (float*)d_out;  // 36,864,000 floats

  sinegen_scan<<<B_, 256, 0, stream>>>(f0, FP);
  sinegen_main<<<NBLK_, 256, 0, stream>>>(FP, rand_ini, noise, out);

  (void)in_sizes; (void)n_in; (void)out_size; (void)ws_size;
}